// stereo_network_30021821399647
// MI455X (gfx1250) — compile-verified
//
#include <hip/hip_runtime.h>

#define AS1 __attribute__((address_space(1)))
#define AS3 __attribute__((address_space(3)))

// 16-byte vector type matching the async-to-LDS builtin's pointee type
// (diagnostic showed: 'int __attribute__((__vector_size__(4 * sizeof(int)))) __device__ *')
typedef int v4i_vs __attribute__((__vector_size__(16)));

// CDNA5 async global->LDS path (gfx1250). Guarded so the host pass and any
// toolchain without the builtin still compile via the plain-LDS fallback.
#if defined(__gfx1250__) && __has_builtin(__builtin_amdgcn_global_load_async_to_lds_b128) && __has_builtin(__builtin_amdgcn_s_wait_asynccnt)
#define USE_ASYNC_LDS 1
#else
#define USE_ASYNC_LDS 0
#endif

struct F3 { float x, y, z; };   // align(4) -> global_store_b96-eligible

// ---------------------------------------------------------------------------
// Kernel 1: one thread per box. All per-box scalar geometry, plus the small
// outputs (proposals_left/right [R,N,5], depth_bin [N,R], depth [N]).
// Packs a 40-float (160B, 16B-aligned) param block per box into d_ws for the
// voxel kernel: center(3), lo(3), hi(3), M(6), P2(12), P3(12), pad(1).
// ---------------------------------------------------------------------------
__global__ void box_precompute(const float* __restrict__ lb,
                               const float* __restrict__ rb,
                               const float* __restrict__ p2s,
                               const float* __restrict__ p3s,
                               const float* __restrict__ fbs,
                               const float* __restrict__ trans,
                               const float* __restrict__ trans_invs,
                               const int*   __restrict__ pR,
                               float* __restrict__ out,
                               float* __restrict__ ws,
                               int N)
{
    const int n = blockIdx.x * blockDim.x + threadIdx.x;
    if (n >= N) return;
    const int R = *pR;

    const float l0 = lb[(size_t)n*5+0], l1 = lb[(size_t)n*5+1], l2 = lb[(size_t)n*5+2];
    const float l3 = lb[(size_t)n*5+3], l4 = lb[(size_t)n*5+4];
    const float r1 = rb[(size_t)n*5+1], r2 = rb[(size_t)n*5+2];
    const float r3 = rb[(size_t)n*5+3], r4 = rb[(size_t)n*5+4];

    const int b = (int)l0;
    const float* Mi = trans_invs + (size_t)b*6;  // feature -> original
    const float* M  = trans      + (size_t)b*6;  // original -> feature
    const float* P2 = p2s        + (size_t)b*12;
    const float* P3 = p3s        + (size_t)b*12;
    const float fb  = fbs[b];

    const float mi0=Mi[0], mi1=Mi[1], mi2=Mi[2], mi3=Mi[3], mi4=Mi[4], mi5=Mi[5];
    const float lb1x = mi0*l1 + mi1*l2 + mi2;
    const float lb1y = mi3*l1 + mi4*l2 + mi5;
    const float lb2x = mi0*l3 + mi1*l4 + mi2;
    const float lb2y = mi3*l3 + mi4*l4 + mi5;
    const float rb1x = mi0*r1 + mi1*r2 + mi2;
    const float rb2x = mi0*r3 + mi1*r4 + mi2;

    const float cx_l = 0.5f*(lb1x + lb2x);
    const float cy_l = 0.5f*(lb1y + lb2y);
    const float cx_r = 0.5f*(rb1x + rb2x);
    const float disp = cx_l - cx_r;
    const float depth = fb / disp;
    const float d_min = fminf(fmaxf(depth - 12.5f, 1.0f), 90.0f);
    const float d_max = fminf(fmaxf(depth + 12.5f, 1.0f), 90.0f);

    // --- output partitions (order of the reference's return tuple) ---
    const size_t NR5 = (size_t)R * N * 5;
    float* propL = out;
    float* propR = out + NR5;
    float* dbin  = out + 2*NR5;
    float* depth_out = out + 2*NR5 + (size_t)N*R + (size_t)N*10000;

    const float inv_rm1 = 1.0f / (float)(R - 1);
    for (int r = 0; r < R; ++r) {
        const float rate = (float)r * inv_rm1;
        const float db = d_max - (d_max - d_min) * rate;
        dbin[(size_t)n*R + r] = db;
        const float dispb = fb / db * 0.25f;
        const float x1s = fmaxf(l1 - dispb, 0.0f);
        const float x2s = fmaxf(l3 - dispb, 0.0f);
        const size_t o = ((size_t)r*N + n) * 5;
        propL[o+0]=l0; propL[o+1]=l1;  propL[o+2]=l2; propL[o+3]=l3;  propL[o+4]=l4;
        propR[o+0]=l0; propR[o+1]=x1s; propR[o+2]=l2; propR[o+3]=x2s; propR[o+4]=l4;
    }
    depth_out[n] = depth;

    // voxel-grid center in rect coords
    const float z = depth - P2[11];
    const float x = (cx_l*depth - P2[3] - P2[2]*z) / P2[0];
    const float y = (cy_l*depth - P2[7] - P2[6]*z) / P2[5];

    float* w = ws + (size_t)n * 40;
    w[0]=x;  w[1]=y;  w[2]=z;
    w[3]=l1; w[4]=l2; w[5]=d_min;      // lo
    w[6]=l3; w[7]=l4; w[8]=d_max;      // hi
    w[9]=M[0]; w[10]=M[1]; w[11]=M[2]; w[12]=M[3]; w[13]=M[4]; w[14]=M[5];
    #pragma unroll
    for (int i = 0; i < 12; ++i) w[15+i] = P2[i];
    #pragma unroll
    for (int i = 0; i < 12; ++i) w[27+i] = P3[i];
    w[39] = 0.0f;
}

// ---------------------------------------------------------------------------
// Kernel 2: one block per box, 256 threads (8 wave32 waves), 1000 voxel cells.
// Param block broadcast via CDNA5 async global->LDS (10 lanes x b128), then
// pure coalesced vector stores (the kernel's real cost: ~330 MB of writes).
// ---------------------------------------------------------------------------
__global__ void voxel_kernel(const float* __restrict__ ws,
                             const int*   __restrict__ pR,
                             float* __restrict__ out,
                             int N)
{
    const int n   = blockIdx.x;
    const int tid = threadIdx.x;
    __shared__ __align__(16) float sp[40];

#if USE_ASYNC_LDS
    if (tid < 10) {
        float* g = const_cast<float*>(ws) + (size_t)n*40 + tid*4;
        __builtin_amdgcn_global_load_async_to_lds_b128(
            (AS1 v4i_vs*)g, (AS3 v4i_vs*)(&sp[tid*4]), /*offset=*/0, /*cpol=*/0);
    }
    __builtin_amdgcn_s_wait_asynccnt(0);
    __syncthreads();
#else
    if (tid < 40) sp[tid] = ws[(size_t)n*40 + tid];
    __syncthreads();
#endif

    const int R = *pR;

    // hoist all params to registers (loop-invariant)
    const float cx = sp[0], cy = sp[1], cz = sp[2];
    const float lo0 = sp[3], lo1 = sp[4], lo2 = sp[5];
    const float i0 = 2.0f / (sp[6] - sp[3]);
    const float i1 = 2.0f / (sp[7] - sp[4]);
    const float i2 = 2.0f / (sp[8] - sp[5]);
    const float m0=sp[9], m1=sp[10], m2=sp[11], m3=sp[12], m4=sp[13], m5=sp[14];
    const float a0=sp[15], a1=sp[16], a2 =sp[17], a3 =sp[18];
    const float a4=sp[19], a5=sp[20], a6 =sp[21], a7 =sp[22];
    const float a8=sp[23], a9=sp[24], a10=sp[25], a11=sp[26];
    const float b0=sp[27], b1=sp[28], b2 =sp[29], b3 =sp[30];
    const float b4=sp[31], b5=sp[32], b6 =sp[33], b7 =sp[34];
    const float b8=sp[35], b9=sp[36], b10=sp[37], b11=sp[38];

    // output partitions
    const size_t NR5 = (size_t)R * N * 5;
    float* nci = out + 2*NR5 + (size_t)N*R;       // [N,1000,3]
    float* val = nci + (size_t)N*3000;            // [N,1000]
    float* nl2 = val + (size_t)N*1000;            // [N,1000,2]
    float* vll = nl2 + (size_t)N*2000;            // [N,1000]
    float* nr2 = vll + (size_t)N*1000;            // [N,1000,2]
    float* vrr = nr2 + (size_t)N*2000;            // [N,1000]

    const float BX = 2.0f / 319.0f;   // INPUT_W/4 - 1
    const float BY = 2.0f / 95.0f;    // INPUT_H/4 - 1

    for (int vi = tid; vi < 1000; vi += blockDim.x) {
        const int ixq = vi / 100;
        const int rem = vi - ixq*100;
        const int iyq = rem / 10;
        const int izq = rem - iyq*10;
        const float crx = cx + (0.5f*(float)ixq - 2.25f);
        const float cry = cy + (0.5f*(float)iyq - 2.25f);
        const float crz = cz + ((float)izq - 4.5f);

        // left projection (P2) + affine M
        const float pw = a8*crx + a9*cry + a10*crz + a11;
        const float px = a0*crx + a1*cry + a2 *crz + a3;
        const float py = a4*crx + a5*cry + a6 *crz + a7;
        const float ul = px / pw, vl = py / pw;
        const float ilx = m0*ul + m1*vl + m2;
        const float ily = m3*ul + m4*vl + m5;

        // right projection (P3) + affine M
        const float qw = b8*crx + b9*cry + b10*crz + b11;
        const float qx = b0*crx + b1*cry + b2 *crz + b3;
        const float qy = b4*crx + b5*cry + b6 *crz + b7;
        const float ur = qx / qw, vr = qy / qw;
        const float irx = m0*ur + m1*vr + m2;
        const float iry = m3*ur + m4*vr + m5;

        // normalized 3D coord + validity
        const float n0 = (ilx - lo0) * i0 - 1.0f;
        const float n1 = (ily - lo1) * i1 - 1.0f;
        const float n2 = (crz - lo2) * i2 - 1.0f;
        const float v3 = (n0 >= -1.0f && n0 <= 1.0f &&
                          n1 >= -1.0f && n1 <= 1.0f &&
                          n2 >= -1.0f && n2 <= 1.0f) ? 1.0f : 0.0f;

        const float nlx = ilx*BX - 1.0f, nly = ily*BY - 1.0f;
        const float vL = (nlx >= -1.0f && nlx <= 1.0f &&
                          nly >= -1.0f && nly <= 1.0f) ? 1.0f : 0.0f;
        const float nrx = irx*BX - 1.0f, nry = iry*BY - 1.0f;
        const float vR = (nrx >= -1.0f && nrx <= 1.0f &&
                          nry >= -1.0f && nry <= 1.0f) ? 1.0f : 0.0f;

        const size_t cell = (size_t)n*1000 + (size_t)vi;
        F3 t; t.x = n0; t.y = n1; t.z = n2;
        *reinterpret_cast<F3*>(nci + cell*3) = t;          // b96
        val[cell] = v3;
        *reinterpret_cast<float2*>(nl2 + cell*2) = make_float2(nlx, nly);  // b64
        vll[cell] = vL;
        *reinterpret_cast<float2*>(nr2 + cell*2) = make_float2(nrx, nry);  // b64
        vrr[cell] = vR;
    }
}

extern "C" void kernel_launch(void* const* d_in, const int* in_sizes, int n_in,
                              void* d_out, int out_size, void* d_ws, size_t ws_size,
                              hipStream_t stream)
{
    (void)n_in; (void)out_size; (void)ws_size;
    const float* lb  = (const float*)d_in[0];
    const float* rb  = (const float*)d_in[1];
    const float* p2s = (const float*)d_in[2];
    const float* p3s = (const float*)d_in[3];
    const float* fbs = (const float*)d_in[4];
    const float* tr  = (const float*)d_in[5];   // trans (orig -> feature)
    const float* tri = (const float*)d_in[6];   // trans_invs (feature -> orig)
    const int*   pR  = (const int*)d_in[7];
    float* out = (float*)d_out;
    float* ws  = (float*)d_ws;    // 40 floats per box (160B, 16B aligned)

    const int N = in_sizes[0] / 5;

    box_precompute<<<(N + 255) / 256, 256, 0, stream>>>(
        lb, rb, p2s, p3s, fbs, tr, tri, pR, out, ws, N);
    voxel_kernel<<<N, 256, 0, stream>>>(ws, pR, out, N);
}